// SparseModel_87900800680513
// MI455X (gfx1250) — compile-verified
//
#include <hip/hip_runtime.h>

typedef __attribute__((ext_vector_type(2))) float v2f;
typedef __attribute__((ext_vector_type(8))) float v8f;

#define B_   16384
#define IN_  512
#define H_   1024
#define OUT_ 10
#define L_   8
#define NW_  16

// wave32 butterfly reductions (5 shfl_xor steps, no barriers)
__device__ __forceinline__ float wave_max(float v) {
#pragma unroll
  for (int i = 16; i > 0; i >>= 1) v = fmaxf(v, __shfl_xor(v, i, 32));
  return v;
}
__device__ __forceinline__ float wave_sum(float v) {
#pragma unroll
  for (int i = 16; i > 0; i >>= 1) v += __shfl_xor(v, i, 32);
  return v;
}

// One workgroup = 128 threads (4 wave32) = 16 rows of the batch.
// Phase A: C[16,1024] = x[16,512] @ W1 + b1 via V_WMMA_F32_16X16X4_F32 -> LDS
// Phase B: 8 layers x 16 softmax-steps entirely in VGPRs (1 row = 32 f32/lane)
// Phase C: logits = x @ Wout + bout, 10-wide softmax, store.
__global__ __launch_bounds__(128) void sparse_model_fused(
    const float* __restrict__ x,    const float* __restrict__ W1,
    const float* __restrict__ b1,   const float* __restrict__ wp,
    const float* __restrict__ wsv,  const float* __restrict__ Wout,
    const float* __restrict__ bout, float* __restrict__ out) {
  __shared__ float lds[24576];           // 96 KB
  float* lds_x = lds;                    // [16 x 512] A tile (32 KB)
  float* lds_c = lds + 16 * IN_;         // [16 x 1024] activations (64 KB)

  const int tid  = threadIdx.x;
  const int wave = tid >> 5;
  const int lane = tid & 31;
  const int r0   = blockIdx.x * 16;

  // ---- stage A tile (16x512) into LDS with 128-bit loads ----
  {
    const float4* src = (const float4*)(x + (size_t)r0 * IN_);
    float4* dst = (float4*)lds_x;
    for (int i = tid; i < 16 * IN_ / 4; i += 128) dst[i] = src[i];
  }
  __syncthreads();

  // ---- Phase A: WMMA fp32 GEMM, each wave owns 16 of the 64 N-tiles ----
  const int mm = lane & 15;              // M (A) / N (B,C) index within tile
  const int kp = (lane >> 4) << 1;       // K sub-pair select (0 or 2)
  for (int t = 0; t < 16; ++t) {
    const int n0 = (wave * 16 + t) * 16;
    const float bv = b1[n0 + mm];
    v8f acc = {bv, bv, bv, bv, bv, bv, bv, bv};
#pragma unroll 4
    for (int k = 0; k < IN_; k += 4) {
      v2f a, b;
      a.x = lds_x[mm * IN_ + k + kp];
      a.y = lds_x[mm * IN_ + k + kp + 1];
      b.x = W1[(size_t)(k + kp)     * H_ + n0 + mm];
      b.y = W1[(size_t)(k + kp + 1) * H_ + n0 + mm];
      acc = __builtin_amdgcn_wmma_f32_16x16x4_f32(
          false, a, false, b, (short)0, acc, false, false);
    }
#pragma unroll
    for (int v = 0; v < 8; ++v)
      lds_c[(size_t)(v + ((lane >> 4) << 3)) * H_ + n0 + mm] = acc[v];
  }
  __syncthreads();

  // ---- Phase B + C: each wave processes 4 rows, row resident in VGPRs ----
  for (int rr = 0; rr < 4; ++rr) {
    const int r = wave * 4 + rr;
    float xv[32];
#pragma unroll
    for (int j = 0; j < 32; ++j) xv[j] = lds_c[(size_t)r * H_ + lane + 32 * j];

    for (int l = 0; l < L_; ++l) {
      // TEMP == 1.0 -> x / TEMP is identity
      for (int s = 0; s < NW_; ++s) {
        const float* wpp = wp + (size_t)(l * NW_ + s) * H_;
        const float  w   = wsv[l * NW_ + s];
        float tt[32];
        float m = -3.0e38f;
#pragma unroll
        for (int j = 0; j < 32; ++j) {
          tt[j] = wpp[lane + 32 * j] * xv[j];
          m = fmaxf(m, tt[j]);
        }
        m = wave_max(m);
        float sum = 0.f;
#pragma unroll
        for (int j = 0; j < 32; ++j) {
          tt[j] = __expf(tt[j] - m);
          sum += tt[j];
        }
        sum = wave_sum(sum);
        const float scale = w / sum;     // x += w * softmax(wp*x)
#pragma unroll
        for (int j = 0; j < 32; ++j) xv[j] += scale * tt[j];
      }
#pragma unroll
      for (int j = 0; j < 32; ++j) xv[j] = fmaxf(xv[j], 0.f);  // relu(relu(x))
    }

    // ---- Phase C: output layer + softmax over 10 ----
    float lg[OUT_];
#pragma unroll
    for (int o = 0; o < OUT_; ++o) lg[o] = 0.f;
#pragma unroll
    for (int j = 0; j < 32; ++j) {
      const float* wrow = Wout + (size_t)(lane + 32 * j) * OUT_;
      const float xvj = xv[j];
#pragma unroll
      for (int o = 0; o < OUT_; ++o) lg[o] += xvj * wrow[o];
    }
#pragma unroll
    for (int o = 0; o < OUT_; ++o) lg[o] = wave_sum(lg[o]) + bout[o];

    float m = lg[0];
#pragma unroll
    for (int o = 1; o < OUT_; ++o) m = fmaxf(m, lg[o]);
    float sum = 0.f;
#pragma unroll
    for (int o = 0; o < OUT_; ++o) { lg[o] = __expf(lg[o] - m); sum += lg[o]; }
    const float inv = 1.f / sum;
    if (lane == 0) {
#pragma unroll
      for (int o = 0; o < OUT_; ++o)
        out[(size_t)(r0 + r) * OUT_ + o] = lg[o] * inv;
    }
  }
}

extern "C" void kernel_launch(void* const* d_in, const int* in_sizes, int n_in,
                              void* d_out, int out_size, void* d_ws, size_t ws_size,
                              hipStream_t stream) {
  (void)in_sizes; (void)n_in; (void)out_size; (void)d_ws; (void)ws_size;
  const float* x    = (const float*)d_in[0];
  const float* W1   = (const float*)d_in[1];
  const float* b1   = (const float*)d_in[2];
  const float* wp   = (const float*)d_in[3];
  const float* wsv  = (const float*)d_in[4];
  const float* Wout = (const float*)d_in[5];
  const float* bout = (const float*)d_in[6];
  float* out = (float*)d_out;

  dim3 grid(B_ / 16), block(128);
  hipLaunchKernelGGL(sparse_model_fused, grid, block, 0, stream,
                     x, W1, b1, wp, wsv, Wout, bout, out);
}